// VectorQuantizer_22746146799642
// MI455X (gfx1250) — compile-verified
//
#include <hip/hip_runtime.h>

typedef __attribute__((ext_vector_type(2))) float v2f;
typedef __attribute__((ext_vector_type(4))) float v4f;
typedef __attribute__((ext_vector_type(8))) float v8f;

#define VQ_D      128
#define VQ_K      1024
#define VQ_N      65536                    // 64*1024 rows
#define VQ_QELEMS (VQ_N * VQ_D)            // 8388608
#define VQ_IDXOFF VQ_QELEMS                // indices start
#define VQ_LOSSOFF (VQ_QELEMS + VQ_N)      // single loss scalar
#define NTILES    (VQ_K / 16)              // 64 codeword tiles
#define LDS_STRIDE 132                     // 128 + 4 pad dwords: conflict-free

// ---------------------------------------------------------------------------
// Kernel 1: cnorm[c] = 0.5 * ||codeword_c||^2 ; also zero the loss slot.
// ---------------------------------------------------------------------------
__global__ void vq_cnorm_kernel(const float* __restrict__ C,
                                float* __restrict__ cnorm,
                                float* __restrict__ loss_slot) {
    int c = blockIdx.x;
    int lane = threadIdx.x;  // 0..31
    const float4* row = (const float4*)(C + c * VQ_D);
    float4 v = row[lane];
    float s = v.x * v.x + v.y * v.y + v.z * v.z + v.w * v.w;
#pragma unroll
    for (int m = 16; m >= 1; m >>= 1) s += __shfl_xor(s, m, 32);
    if (lane == 0) {
        cnorm[c] = 0.5f * s;
        if (c == 0) *loss_slot = 0.0f;
    }
}

// ---------------------------------------------------------------------------
// Kernel 2: 256 threads = 8 waves; each wave owns a 16-row tile of X.
// Block cooperatively double-buffers 16-codeword B tiles through LDS.
// FP32 WMMA GEMM; per-lane running argmin in the loop, one cross-lane
// reduction at the end; then gather + indices + loss.
// ---------------------------------------------------------------------------
__global__ void __launch_bounds__(256) vq_main_kernel(
    const float* __restrict__ X, const float* __restrict__ C,
    const float* __restrict__ cnorm, float* __restrict__ out) {
    __shared__ float smem[2][16 * LDS_STRIDE];   // 2 x 8.25 KB

    const int tid    = threadIdx.x;
    const int lane   = tid & 31;
    const int wib    = tid >> 5;                 // wave in block (0..7)
    const int rowBase = (blockIdx.x * 8 + wib) * 16;
    const int lane15 = lane & 15;
    const int hi     = lane >> 4;                // half-wave id
    const int dOff   = hi * 2;                   // K offset within 4-chunk

    // Cooperative staging coords: 256 threads x 8 floats = 16x128 tile.
    const int scw = tid >> 4;                    // codeword row in tile (0..15)
    const int sf  = (tid & 15) * 8;              // float offset in row

    // ---- A fragments (16 rows x 128), ISA 16x4 f32 A layout:
    // lanes 0-15: row=lane, K={4k,4k+1}; lanes 16-31: row=lane-16, K={4k+2,4k+3}.
    v2f a[32];
    const float* xrow = X + (size_t)(rowBase + lane15) * VQ_D + dOff;
#pragma unroll
    for (int kk = 0; kk < 32; ++kk)
        a[kk] = *(const v2f*)(xrow + kk * 4);

    // ---- Prologue: stage tile 0 into smem[0] (coalesced b128 loads).
    {
        const float* src = C + (size_t)scw * VQ_D + sf;
        v4f r0 = *(const v4f*)(src);
        v4f r1 = *(const v4f*)(src + 4);
        float* dst = &smem[0][scw * LDS_STRIDE + sf];
        *(v4f*)(dst)     = r0;
        *(v4f*)(dst + 4) = r1;
    }

    // Per-lane running min over this lane's column class (col % 16 == lane15).
    float bestVal[8];
    int   bestIdx[8];
#pragma unroll
    for (int r = 0; r < 8; ++r) { bestVal[r] = 3.4028235e38f; bestIdx[r] = 0; }

    // ---- Sweep 64 codeword tiles (ascending ct: first-min tie-break kept).
    for (int ct = 0; ct < NTILES; ++ct) {
        __syncthreads();  // smem[ct&1] ready; smem[(ct+1)&1] free to overwrite

        // Issue next tile's global loads early (held in regs, overlap compute)
        const bool hasNext = (ct + 1 < NTILES);
        v4f n0, n1;
        if (hasNext) {
            const float* src = C + (size_t)((ct + 1) * 16 + scw) * VQ_D + sf;
            n0 = *(const v4f*)(src);
            n1 = *(const v4f*)(src + 4);
        }

        // Prefetch all 32 B fragments from LDS (conflict-free: stride 132),
        // then run the WMMA chain over D=128.
        const float* bbase = &smem[ct & 1][lane15 * LDS_STRIDE + dOff];
        v2f b[32];
#pragma unroll
        for (int kk = 0; kk < 32; ++kk)
            b[kk] = *(const v2f*)(bbase + kk * 4);

        v8f acc = {};
#pragma unroll
        for (int kk = 0; kk < 32; ++kk)
            acc = __builtin_amdgcn_wmma_f32_16x16x4_f32(
                false, a[kk], false, b[kk], (short)0, acc, false, false);

        // Per-lane running argmin of 0.5||c||^2 - x.c (no cross-lane work
        // inside the hot loop). C/D layout: lane l, vgpr r -> row r+8*hi,
        // col = ct*16 + (l&15).
        const int col = ct * 16 + lane15;
        const float cn = cnorm[col];
#pragma unroll
        for (int r = 0; r < 8; ++r) {
            float v = cn - acc[r];
            if (v < bestVal[r]) { bestVal[r] = v; bestIdx[r] = col; }
        }

        // Commit staged registers into the other LDS buffer.
        if (hasNext) {
            float* dst = &smem[(ct + 1) & 1][scw * LDS_STRIDE + sf];
            *(v4f*)(dst)     = n0;
            *(v4f*)(dst + 4) = n1;
        }
    }

    // ---- One cross-lane (16-lane, per half-wave) value+index reduction.
    // Ties: smaller index wins == jnp.argmin first-min semantics.
#pragma unroll
    for (int r = 0; r < 8; ++r) {
        float v = bestVal[r];
        int   i = bestIdx[r];
#pragma unroll
        for (int m = 1; m < 16; m <<= 1) {
            float ov = __shfl_xor(v, m, 32);
            int   oi = __shfl_xor(i, m, 32);
            if (ov < v || (ov == v && oi < i)) { v = ov; i = oi; }
        }
        bestVal[r] = v;
        bestIdx[r] = i;   // uniform across each half-wave now
    }

    // ---- Encoding indices (reference int32 flattened to float output).
    if (lane15 == 0) {
#pragma unroll
        for (int r = 0; r < 8; ++r) {
            int row = rowBase + hi * 8 + r;
            out[VQ_IDXOFF + row] = (float)bestIdx[r];
        }
    }

    // ---- Gather quantized = codewords[idx]; accumulate loss directly
    // from (q - x)^2 for numerical fidelity (X re-read hits L2).
    float err = 0.0f;
#pragma unroll
    for (int r = 0; r < 8; ++r) {
        const int row = rowBase + hi * 8 + r;
        const int idx = bestIdx[r];              // uniform within half-wave
        const float* cq = C + (size_t)idx * VQ_D;
        const float* xr = X + (size_t)row * VQ_D;
        float* qr = out + (size_t)row * VQ_D;
#pragma unroll
        for (int d0 = 0; d0 < 4; ++d0) {
            int d = d0 * 32 + lane15 * 2;        // coalesced per half-wave
            v2f q = *(const v2f*)(cq + d);
            v2f x = *(const v2f*)(xr + d);
            float ex = q.x - x.x, ey = q.y - x.y;
            err += ex * ex + ey * ey;
            *(v2f*)(qr + d) = q;
        }
    }
#pragma unroll
    for (int m = 16; m >= 1; m >>= 1) err += __shfl_xor(err, m, 32);
    if (lane == 0) {
        // loss = (1 + 0.25) * mean((q - x)^2)
        atomicAdd(out + VQ_LOSSOFF, err * (1.25f / (float)VQ_QELEMS));
    }
}

// ---------------------------------------------------------------------------
extern "C" void kernel_launch(void* const* d_in, const int* in_sizes, int n_in,
                              void* d_out, int out_size, void* d_ws, size_t ws_size,
                              hipStream_t stream) {
    (void)in_sizes; (void)n_in; (void)out_size; (void)ws_size;
    const float* X = (const float*)d_in[0];   // (64,1024,128) f32
    const float* C = (const float*)d_in[1];   // (1024,128) f32
    float* out   = (float*)d_out;             // [quantized | indices | loss]
    float* cnorm = (float*)d_ws;              // 1024 floats scratch

    vq_cnorm_kernel<<<VQ_K, 32, 0, stream>>>(C, cnorm, out + VQ_LOSSOFF);

    const int waves  = VQ_N / 16;             // 4096 row tiles
    const int blocks = waves / 8;             // 512 blocks of 8 waves
    vq_main_kernel<<<blocks, 256, 0, stream>>>(X, C, cnorm, out);
}